// GroupEncoder_22806276342098
// MI455X (gfx1250) — compile-verified
//
#include <hip/hip_runtime.h>
#include <math.h>

typedef float v2f __attribute__((ext_vector_type(2)));
typedef float v4f __attribute__((ext_vector_type(4)));
typedef float v8f __attribute__((ext_vector_type(8)));
typedef int   v4i __attribute__((ext_vector_type(4)));

#define GE_EPS 1e-6f

// ---------------------------------------------------------------- utilities

__device__ __forceinline__ float ge_silu(float v) {
    return v / (1.0f + __expf(-v));
}

__device__ __forceinline__ float ge_softplus(float v) {
    return (v > 20.0f) ? v : log1pf(__expf(v));
}

__device__ __forceinline__ unsigned ge_hash(unsigned a, unsigned b) {
    unsigned h = a * 0x9E3779B9u + (b ^ 0x7F4A7C15u);
    h ^= h >> 16; h *= 0x85EBCA6Bu;
    h ^= h >> 13; h *= 0xC2B2AE35u;
    h ^= h >> 16;
    return h;
}

__device__ __forceinline__ float ge_u01(unsigned h) {
    return ((h >> 8) + 0.5f) * (1.0f / 16777216.0f);   // (0,1)
}

// Marsaglia-Tsang Gamma(alpha, 1) sampler, counter-based RNG keyed on gid.
__device__ float ge_gamma_sample(float alpha, unsigned gid) {
    float a = alpha;
    float boost = 1.0f;
    if (a < 1.0f) {
        float u = ge_u01(ge_hash(gid, 0xB005Fu));
        boost = powf(u, 1.0f / a);
        a += 1.0f;
    }
    float d = a - (1.0f / 3.0f);
    float c = rsqrtf(9.0f * d);
    float res = d; // fallback if rejection never accepts
    for (int it = 0; it < 16; ++it) {
        unsigned h0 = ge_hash(gid * 3u + 1u, (unsigned)(2 * it));
        unsigned h1 = ge_hash(gid * 3u + 2u, (unsigned)(2 * it + 1));
        float u1 = ge_u01(h0);
        float u2 = ge_u01(h1);
        float r  = sqrtf(-2.0f * __logf(u1));
        float xn = r * __cosf(6.28318530718f * u2);   // Box-Muller normal
        float t  = 1.0f + c * xn;
        if (t <= 0.0f) continue;
        float v = t * t * t;
        float u = ge_u01(ge_hash(gid * 3u, (unsigned)it));
        if (__logf(u) < 0.5f * xn * xn + d * (1.0f - v + __logf(v))) {
            res = d * v;
            break;
        }
    }
    return boost * res;
}

// ---------------------------------------------------------------- kernels

__global__ void ge_zero_kernel(float* __restrict__ p, long n) {
    long i = (long)blockIdx.x * blockDim.x + threadIdx.x;
    long stride = (long)gridDim.x * blockDim.x;
    for (; i < n; i += stride) p[i] = 0.0f;
}

// Scatter-add: 16 rows per 256-thread block, 16 lanes per row, 4 features
// per lane via global_load_b128 (streamed non-temporally); accumulation via
// hardware global_atomic_add_f32 (relaxed, agent scope) hitting L2-resident
// accumulators (25.6 MB << 192 MB L2).
__global__ __launch_bounds__(256) void ge_scatter_kernel(
        const float* __restrict__ x,
        const int*   __restrict__ labels,
        float* __restrict__ gsum,
        float* __restrict__ gcnt,
        int B) {
    int row = blockIdx.x * 16 + (threadIdx.x >> 4);
    int q   = threadIdx.x & 15;          // 16-byte quad within the 256 B row
    if (row >= B) return;
    int g = labels[row];
    const v4f* xq = (const v4f*)(x + (long)row * 64) + q;
    v4f v = __builtin_nontemporal_load(xq);
    float* dst = gsum + (long)g * 64 + q * 4;
#pragma unroll
    for (int e = 0; e < 4; ++e) {
        __hip_atomic_fetch_add(&dst[e], v[e],
                               __ATOMIC_RELAXED, __HIP_MEMORY_SCOPE_AGENT);
    }
    if (q == 0) {
        __hip_atomic_fetch_add(&gcnt[g], 1.0f,
                               __ATOMIC_RELAXED, __HIP_MEMORY_SCOPE_AGENT);
    }
}

// One wave32 per 16-group tile. Two 16x64 @ 64x64 GEMM layers via
// V_WMMA_F32_16X16X4_F32, heads + softplus + gamma sample at the end.
#define GE_WAVES_PER_BLOCK 8
__global__ __launch_bounds__(256) void ge_mlp_kernel(
        const float* __restrict__ gsum, const float* __restrict__ gcnt,
        const float* __restrict__ W1,   const float* __restrict__ b1,
        const float* __restrict__ W2,   const float* __restrict__ b2,
        const float* __restrict__ Wa,   const float* __restrict__ ba,
        const float* __restrict__ Wb,   const float* __restrict__ bb,
        float* __restrict__ alpha_out,  float* __restrict__ beta_out,
        float* __restrict__ tau_group,  int G) {
    __shared__ float smem[GE_WAVES_PER_BLOCK][16 * 64];

    int wave = threadIdx.x >> 5;
    int lane = threadIdx.x & 31;
    int tile = blockIdx.x * GE_WAVES_PER_BLOCK + wave;
    int ntiles = (G + 15) >> 4;
    if (tile >= ntiles) return;   // wave-uniform exit

    float* hbuf = smem[wave];

    int half = lane >> 4;         // 0: K pair {0,1}; 1: K pair {2,3} per step
    int mrow = lane & 15;         // A-operand row / C-operand column index
    int g  = tile * 16 + mrow;
    int gc = (g < G) ? g : (G - 1);

    float rcount = 1.0f / fmaxf(gcnt[gc], 1.0f);
    const float* xrow = gsum + (long)gc * 64;

    // ---------------- layer 1: H1 = silu(X @ W1 + b1) ----------------
    v8f acc[4];
#pragma unroll
    for (int n = 0; n < 4; ++n)
#pragma unroll
        for (int e = 0; e < 8; ++e) acc[n][e] = 0.0f;

    for (int k0 = 0; k0 < 64; k0 += 4) {
        int kb = k0 + 2 * half;
        v2f a;
        a.x = xrow[kb + 0] * rcount;
        a.y = xrow[kb + 1] * rcount;
#pragma unroll
        for (int n = 0; n < 4; ++n) {
            int col = n * 16 + mrow;
            v2f bf;
            bf.x = W1[(kb + 0) * 64 + col];
            bf.y = W1[(kb + 1) * 64 + col];
            acc[n] = __builtin_amdgcn_wmma_f32_16x16x4_f32(
                false, a, false, bf, (short)0, acc[n], false, false);
        }
    }
    // C/D layout: VGPR r -> M = r + 8*half, N = n*16 + mrow
#pragma unroll
    for (int n = 0; n < 4; ++n) {
        float bias = b1[n * 16 + mrow];
#pragma unroll
        for (int r = 0; r < 8; ++r) {
            int m = r + 8 * half;
            hbuf[m * 64 + n * 16 + mrow] = ge_silu(acc[n][r] + bias);
        }
    }

    // ---------------- layer 2: H2 = silu(H1 @ W2 + b2) ----------------
#pragma unroll
    for (int n = 0; n < 4; ++n)
#pragma unroll
        for (int e = 0; e < 8; ++e) acc[n][e] = 0.0f;

    for (int k0 = 0; k0 < 64; k0 += 4) {
        int kb = k0 + 2 * half;
        v2f a;
        a.x = hbuf[mrow * 64 + kb + 0];
        a.y = hbuf[mrow * 64 + kb + 1];
#pragma unroll
        for (int n = 0; n < 4; ++n) {
            int col = n * 16 + mrow;
            v2f bf;
            bf.x = W2[(kb + 0) * 64 + col];
            bf.y = W2[(kb + 1) * 64 + col];
            acc[n] = __builtin_amdgcn_wmma_f32_16x16x4_f32(
                false, a, false, bf, (short)0, acc[n], false, false);
        }
    }
#pragma unroll
    for (int n = 0; n < 4; ++n) {
        float bias = b2[n * 16 + mrow];
#pragma unroll
        for (int r = 0; r < 8; ++r) {
            int m = r + 8 * half;
            hbuf[m * 64 + n * 16 + mrow] = ge_silu(acc[n][r] + bias);
        }
    }

    // ---------------- heads + gamma sample (lanes 0..15 = one group each)
    if (lane < 16 && g < G) {
        float sa = 0.0f, sb = 0.0f;
        for (int h = 0; h < 64; ++h) {
            float hv = hbuf[mrow * 64 + h];
            sa += hv * Wa[h];
            sb += hv * Wb[h];
        }
        float alpha = ge_softplus(sa + ba[0]) + GE_EPS;
        float beta  = ge_softplus(sb + bb[0]) + GE_EPS;
        alpha_out[g] = alpha;
        beta_out[g]  = beta;
        float gsamp  = ge_gamma_sample(alpha, (unsigned)g);
        tau_group[g] = gsamp / beta;
    }
}

// Gather tau back per reflection, 4 elements per thread:
// b128 label load + 4 gathered b32 loads (L2-resident table) + b128 store.
__global__ __launch_bounds__(256) void ge_gather_kernel(
        const int* __restrict__ labels,
        const float* __restrict__ tau_group,
        float* __restrict__ tau_out, int B) {
    int i4 = blockIdx.x * blockDim.x + threadIdx.x;   // quad index
    int base = i4 * 4;
    if (base + 3 < B) {
        v4i l = *(const v4i*)(labels + base);
        v4f t;
#pragma unroll
        for (int e = 0; e < 4; ++e) t[e] = tau_group[l[e]];
        __builtin_nontemporal_store(t, (v4f*)(tau_out + base));
    } else {
        for (int b = base; b < B; ++b) tau_out[b] = tau_group[labels[b]];
    }
}

// ---------------------------------------------------------------- launch

extern "C" void kernel_launch(void* const* d_in, const int* in_sizes, int n_in,
                              void* d_out, int out_size, void* d_ws, size_t ws_size,
                              hipStream_t stream) {
    const float* x      = (const float*)d_in[0];
    const int*   labels = (const int*)  d_in[1];
    // d_in[2] = n_groups (device scalar) -- derived from sizes instead
    const float* W1 = (const float*)d_in[3];
    const float* b1 = (const float*)d_in[4];
    const float* W2 = (const float*)d_in[5];
    const float* b2 = (const float*)d_in[6];
    const float* Wa = (const float*)d_in[7];
    const float* ba = (const float*)d_in[8];
    const float* Wb = (const float*)d_in[9];
    const float* bb = (const float*)d_in[10];

    int B = in_sizes[1];              // 2,097,152
    int G = (out_size - B) / 2;       // 100,000

    float* out       = (float*)d_out;
    float* alpha_out = out;
    float* beta_out  = out + G;
    float* tau_out   = out + 2L * G;

    float* gsum = (float*)d_ws;              // G*64 floats (25.6 MB)
    float* gcnt = gsum + (long)G * 64;       // G floats
    float* taug = gcnt + G;                  // G floats

    long zcount = (long)G * 64 + 2L * G;
    ge_zero_kernel<<<1024, 256, 0, stream>>>(gsum, zcount);

    ge_scatter_kernel<<<(B + 15) / 16, 256, 0, stream>>>(x, labels, gsum, gcnt, B);

    int ntiles = (G + 15) / 16;
    int nblocks = (ntiles + GE_WAVES_PER_BLOCK - 1) / GE_WAVES_PER_BLOCK;
    ge_mlp_kernel<<<nblocks, 256, 0, stream>>>(gsum, gcnt, W1, b1, W2, b2,
                                               Wa, ba, Wb, bb,
                                               alpha_out, beta_out, taug, G);

    int quads = (B + 3) / 4;
    ge_gather_kernel<<<(quads + 255) / 256, 256, 0, stream>>>(labels, taug, tau_out, B);
}